// FeatureExtractor_42554535969194
// MI455X (gfx1250) — compile-verified
//
#include <hip/hip_runtime.h>
#include <math.h>

// ---------------- problem constants (match reference) ----------------
#define NFFT   512
#define HOP    160
#define NMELS  64
#define NMFCC  13
#define NFREQ  257
#define KPAD   260          // 257 padded up to multiple of 4 for WMMA K
#define BATCH  32
#define NSAMP  320000
#define TFRM   2000         // 1 + (320000+352-512)/160
#define TEN    1997         // 1 + (320000-512)/160
#define PADL   (NFFT - HOP) // 352 left pad before framing
#define PREEMPH 0.95f

typedef float v2f __attribute__((ext_vector_type(2)));
typedef float v8f __attribute__((ext_vector_type(8)));

// =====================================================================
// Kernel 1: preemphasis + framing + window + 512-pt FFT (packed pairs)
//           + power spectrum + mel GEMM (f32 WMMA) + log + DCT GEMM
// One block = one batch element x 16 consecutive frames.
// =====================================================================
__global__ __launch_bounds__(256)
void mfcc_kernel(const float* __restrict__ sig,
                 const float* __restrict__ mel,   // [257][64]
                 const float* __restrict__ dct,   // [13][64]
                 const float* __restrict__ win,   // [512]
                 float* __restrict__ out_logmel,  // [B][T][64]
                 float* __restrict__ out_mfcc)    // [B][T][13]
{
    __shared__ float twr[256], twi[256];        // W_512^k twiddles
    __shared__ float zr[8][NFFT], zi[8][NFFT];  // 8 packed complex FFTs
    __shared__ float pw[16][KPAD];              // power spectra (zero-padded K)
    __shared__ float lm[16][NMELS];             // log-mel tile

    const int tid = threadIdx.x;
    const int b   = blockIdx.y;
    const int t0  = blockIdx.x * 16;
    const float* x = sig + (size_t)b * NSAMP;

    // ---- twiddle table: exp(-2*pi*i*k/512), k = 0..255 ----
    {
        float ang = -6.28318530717958647692f * (float)tid / (float)NFFT;
        twr[tid] = cosf(ang);
        twi[tid] = sinf(ang);
    }

    // ---- load 16 frames: preemphasis, left-pad 352, window, pack two
    //      real frames per complex sequence, bit-reversed store ----
    for (int f = 0; f < 8; ++f) {
        const int tp = t0 + 2 * f;              // even frame of the pair
        for (int hh = 0; hh < 2; ++hh) {
            const int n  = tid + hh * 256;      // 0..511 within frame
            const float wv = win[n];
            const int jp = tp * HOP + n - PADL; // index into preemphasized sig
            float sp = 0.0f;
            if (jp == 0)      sp = x[0];
            else if (jp > 0)  sp = x[jp] - PREEMPH * x[jp - 1];
            const int jq = jp + HOP;            // odd frame of the pair
            float sq = 0.0f;
            if (jq == 0)      sq = x[0];
            else if (jq > 0)  sq = x[jq] - PREEMPH * x[jq - 1];
            const int r = (int)(__brev((unsigned)n) >> 23); // 9-bit reverse
            zr[f][r] = sp * wv;
            zi[f][r] = sq * wv;
        }
    }
    __syncthreads();

    // ---- iterative radix-2 DIT FFT, 9 stages, 8 FFTs per stage ----
    for (int len = 2, tstep = 256; len <= NFFT; len <<= 1, tstep >>= 1) {
        const int half = len >> 1;
        const int j  = tid & (half - 1);
        const int p1 = (tid << 1) - j;          // = (tid/half)*len + j
        const int p2 = p1 + half;
        const float wr = twr[j * tstep];
        const float wi = twi[j * tstep];
        for (int f = 0; f < 8; ++f) {
            const float ur = zr[f][p1], ui = zi[f][p1];
            const float xr = zr[f][p2], xi = zi[f][p2];
            const float vr = xr * wr - xi * wi;
            const float vi = xr * wi + xi * wr;
            zr[f][p1] = ur + vr;  zi[f][p1] = ui + vi;
            zr[f][p2] = ur - vr;  zi[f][p2] = ui - vi;
        }
        __syncthreads();
    }

    // ---- unpack two real spectra per packed FFT; power = re^2+im^2 ----
    for (int f = 0; f < 8; ++f) {
        const int k  = tid;                     // 0..255
        const int kk = (NFFT - k) & (NFFT - 1);
        const float ar = zr[f][k],  ai = zi[f][k];
        const float br = zr[f][kk], bi = zi[f][kk];
        const float pr = 0.5f * (ar + br), pi = 0.5f * (ai - bi);
        const float qr = 0.5f * (ai + bi), qi = 0.5f * (br - ar);
        pw[2 * f    ][k] = pr * pr + pi * pi;
        pw[2 * f + 1][k] = qr * qr + qi * qi;
        if (tid == 0) {                         // Nyquist bin k = 256
            const float nr = zr[f][256], ni = zi[f][256];
            pw[2 * f    ][256] = nr * nr;
            pw[2 * f + 1][256] = ni * ni;
        }
    }
    if (tid < 16) { pw[tid][257] = 0.0f; pw[tid][258] = 0.0f; pw[tid][259] = 0.0f; }
    __syncthreads();

    const int wave = tid >> 5;
    const int lane = tid & 31;
    const int hl   = lane >> 4;   // k-half selector (A & B layouts)
    const int ml   = lane & 15;   // row (A) / col (B) selector

    // ---- mel projection: [16 x 260] x [260 x 64] with V_WMMA_F32_16X16X4_F32
    //      waves 0..3 each own one 16-wide mel tile ----
    if (wave < 4) {
        v8f acc = {0.f, 0.f, 0.f, 0.f, 0.f, 0.f, 0.f, 0.f};
        const int nn = wave * 16 + ml;          // mel column 0..63
        for (int kb = 0; kb < KPAD; kb += 4) {
            const int kA = kb + 2 * hl;
            v2f a, bb;
            a.x = pw[ml][kA];
            a.y = pw[ml][kA + 1];
            const int k0 = (kA     <= 256) ? kA     : 256; // pad lanes: a==0
            const int k1 = (kA + 1 <= 256) ? kA + 1 : 256;
            bb.x = mel[k0 * NMELS + nn];
            bb.y = mel[k1 * NMELS + nn];
            acc = __builtin_amdgcn_wmma_f32_16x16x4_f32(
                      false, a, false, bb, (short)0, acc, false, false);
        }
        // log + store (C/D layout: vgpr v -> row v + 8*hl, col = ml)
        for (int v = 0; v < 8; ++v) {
            const int row = v + 8 * hl;
            float mv = acc[v];
            mv = (mv < 1e-8f) ? 1e-8f : mv;
            const float lv = logf(mv);
            lm[row][nn] = lv;
            out_logmel[((size_t)b * TFRM + t0 + row) * NMELS + nn] = lv;
        }
    }
    __syncthreads();

    // ---- MFCC: [16 x 64] x [64 x 13(pad16)] on wave 0 ----
    if (wave == 0) {
        v8f acc = {0.f, 0.f, 0.f, 0.f, 0.f, 0.f, 0.f, 0.f};
        for (int kb = 0; kb < NMELS; kb += 4) {
            const int kA = kb + 2 * hl;
            v2f a, bb;
            a.x = lm[ml][kA];
            a.y = lm[ml][kA + 1];
            bb.x = (ml < NMFCC) ? dct[ml * NMELS + kA    ] : 0.0f;
            bb.y = (ml < NMFCC) ? dct[ml * NMELS + kA + 1] : 0.0f;
            acc = __builtin_amdgcn_wmma_f32_16x16x4_f32(
                      false, a, false, bb, (short)0, acc, false, false);
        }
        for (int v = 0; v < 8; ++v) {
            const int row = v + 8 * hl;
            if (ml < NMFCC)
                out_mfcc[((size_t)b * TFRM + t0 + row) * NMFCC + ml] = acc[v];
        }
    }
}

// =====================================================================
// Kernel 2: per-frame energy features on the raw (un-padded) signal.
// =====================================================================
__global__ __launch_bounds__(256)
void energy_kernel(const float* __restrict__ sig,
                   const float* __restrict__ win,
                   float* __restrict__ out_db,      // [B][TEN][2] (writes ch 0)
                   float* __restrict__ ws_logmax)   // [B][TEN]
{
    const int idx = blockIdx.x * 256 + threadIdx.x;
    if (idx >= BATCH * TEN) return;
    const int b  = idx / TEN;
    const int te = idx % TEN;
    const float* x = sig + (size_t)b * NSAMP + (size_t)te * HOP;

    float s = 0.0f, wsum = 0.0f, mx = 0.0f;
    for (int n = 0; n < NFFT; ++n) {
        const float xv = x[n];
        const float e  = xv * xv;
        const float wv = win[n];
        s    += e * wv;
        wsum += wv;
        mx    = fmaxf(mx, e);
    }
    const float wmean = wsum * (1.0f / NFFT);
    const float meanv = (s / wmean) * (1.0f / NFFT);
    float db = log10f(fmaxf(meanv, 1e-8f));
    db = (fmaxf(db, -8.0f) + 8.0f) * 0.125f;
    out_db[(size_t)idx * 2] = db;

    const float emax = mx * 1.1f;
    ws_logmax[idx] = logf(fmaxf(emax, 1e-8f));
}

// =====================================================================
// Kernel 3: sequential conditional leaky integrator (lax.scan), 1 lane/clip.
// =====================================================================
__global__ void scan_kernel(const float* __restrict__ ws_logmax,
                            float* __restrict__ out_db,   // writes channel 1
                            float* __restrict__ out_mm)   // [B][TEN]
{
    const int b = threadIdx.x;
    if (b >= BATCH) return;
    const float* lmx = ws_logmax + (size_t)b * TEN;
    const float du = 0.997f;                 // 1 - (1-0.999)*3
    float t = lmx[0];                        // scan carry init = first element
    for (int te = 0; te < TEN; ++te) {
        const float cur = lmx[te];
        const float up = t * du + cur * (1.0f - du);
        const float dn = t * du + cur * (1.0f - 0.999f);
        t = (cur > t) ? up : dn;
        float mv = expf(t);
        mv = fminf(fmaxf(mv, 1e-4f), 1.0f);
        out_mm[(size_t)b * TEN + te] = mv;
        out_db[((size_t)b * TEN + te) * 2 + 1] = mv;
    }
}

// =====================================================================
extern "C" void kernel_launch(void* const* d_in, const int* in_sizes, int n_in,
                              void* d_out, int out_size, void* d_ws, size_t ws_size,
                              hipStream_t stream)
{
    const float* sig = (const float*)d_in[0];   // [32][320000]
    const float* mel = (const float*)d_in[1];   // [257][64]
    const float* dct = (const float*)d_in[2];   // [13][64]
    const float* win = (const float*)d_in[3];   // [512]

    float* out        = (float*)d_out;
    float* out_logmel = out;                                        // 32*2000*64
    float* out_mfcc   = out_logmel + (size_t)BATCH * TFRM * NMELS;  // 32*2000*13
    float* out_db     = out_mfcc   + (size_t)BATCH * TFRM * NMFCC;  // 32*1997*2
    float* out_mm     = out_db     + (size_t)BATCH * TEN * 2;       // 32*1997
    float* ws_logmax  = (float*)d_ws;                               // 32*1997 f32

    dim3 g1(TFRM / 16, BATCH);
    mfcc_kernel<<<g1, 256, 0, stream>>>(sig, mel, dct, win, out_logmel, out_mfcc);

    const int nframes = BATCH * TEN;
    energy_kernel<<<(nframes + 255) / 256, 256, 0, stream>>>(sig, win, out_db, ws_logmax);

    scan_kernel<<<1, 32, 0, stream>>>(ws_logmax, out_db, out_mm);
}